// CombinedLoss_29394756173980
// MI455X (gfx1250) — compile-verified
//
#include <hip/hip_runtime.h>
#include <math.h>
#include <stdint.h>

typedef __attribute__((ext_vector_type(2))) float v2f;
typedef __attribute__((ext_vector_type(4))) float v4f;
typedef __attribute__((ext_vector_type(8))) float v8f;
typedef __attribute__((ext_vector_type(4))) int   v4i;

#define B_DIM 256
#define N_DIM 16384
#define P_DIM 128
#define M_TOT (B_DIM * N_DIM)          // 4194304 elements
#define SEG_BLOCKS 1024
#define SEG_THREADS 256
#define CONT_GROUPS 16                 // 256 rows / 16-row stripes
#define CONT_THREADS 256               // 8 waves of 32
#define FIN_THREADS 256

#define TEMP_INV (1.0f / 0.07f)
#define EPSF 1e-6f
#define SMOOTH 1e-6f

// ---- dynamic-LDS layout for the contrastive kernel (float offsets) --------
#define PL        132                      // padded row stride: 528B, 16B-aligned,
                                           // 132 % 64 == 4 -> conflict-free b64 frag reads
#define L_PROJ0   0                        // 256 x 132 staged proj image
#define L_SIM0    (B_DIM * PL)             // 33792: sim stripe 16 x 257
#define SIMSTR    (B_DIM + 1)
#define L_AFF0    (L_SIM0 + 16 * SIMSTR)   // 37904 (ints)
#define L_INST0   (L_AFF0 + B_DIM)         // 38160 (ints)
#define L_PMAX0   (L_INST0 + B_DIM)        // 38416
#define L_PSE0    (L_PMAX0 + 256)          // 38672
#define L_PPS0    (L_PSE0 + 256)           // 38928
#define L_PPC0    (L_PPS0 + 256)           // 39184
#define L_RMAX0   (L_PPC0 + 256)           // 39440
#define L_WL0     (L_RMAX0 + 16)           // 39456
#define L_PS0     (L_WL0 + 16)             // 39472
#define L_CN0     (L_PS0 + 16)             // 39488
#define L_TOTALF  (L_CN0 + 16)             // 39504 floats = 158016 bytes
#define CONT_SMEM_BYTES (L_TOTALF * 4)

#if defined(__HIP_DEVICE_COMPILE__) && __has_builtin(__builtin_amdgcn_global_load_async_to_lds_b128)
#define HAVE_ASYNC_LDS 1
#else
#define HAVE_ASYNC_LDS 0
#endif

// ---------------------------------------------------------------------------
// Kernel 1: segmentation partial reduction (HBM-bound streaming, 33.5 MB).
// Native v_exp/v_log keep VALU work under the 23.3 TB/s roofline time.
// Per block: (sum_focal, sum_p, sum_t, sum_p*t) -> segP[4*block .. +3]
// ---------------------------------------------------------------------------
__global__ void __launch_bounds__(SEG_THREADS)
seg_partial_kernel(const float* __restrict__ logits,
                   const int* __restrict__ gt,
                   float* __restrict__ segP) {
    const int tid = threadIdx.x;
    const int gid = blockIdx.x * SEG_THREADS + tid;
    const int nchunk = M_TOT / 4;      // float4 / int4 chunks
    const int stride = SEG_BLOCKS * SEG_THREADS;

    float fs = 0.0f, sp = 0.0f, st = 0.0f, spt = 0.0f;

    for (int c = gid; c < nchunk; c += stride) {
        v4f x4 = ((const v4f*)logits)[c];
        v4i t4 = ((const v4i*)gt)[c];
#pragma unroll
        for (int k = 0; k < 4; ++k) {
            float x = x4[k];
            float t = (float)t4[k];
            float p = 1.0f / (1.0f + __expf(-x));           // sigmoid (v_exp_f32)
            float pc = fminf(fmaxf(p, EPSF), 1.0f - EPSF);  // clamp for focal
            float omp = 1.0f - pc;
            float focal = (t != 0.0f)
                ? (-0.25f * omp * omp * __logf(pc))
                : (-0.75f * pc  * pc  * __logf(1.0f - pc));
            fs  += focal;
            sp  += p;
            st  += t;
            spt += p * t;
        }
    }

    __shared__ v4f red[SEG_THREADS];
    red[tid] = (v4f){fs, sp, st, spt};
    __syncthreads();
#pragma unroll
    for (int s = SEG_THREADS / 2; s > 0; s >>= 1) {
        if (tid < s) red[tid] += red[tid + s];
        __syncthreads();
    }
    if (tid == 0) {
        v4f r = red[0];
        segP[blockIdx.x * 4 + 0] = r.x;
        segP[blockIdx.x * 4 + 1] = r.y;
        segP[blockIdx.x * 4 + 2] = r.z;
        segP[blockIdx.x * 4 + 3] = r.w;
    }
}

// ---------------------------------------------------------------------------
// Kernel 2: contrastive partials via fp32 WMMA (V_WMMA_F32_16X16X4_F32).
// Stage proj (128 KB) into LDS once per workgroup via async global->LDS
// b128 (ASYNCcnt path), then run the WMMA k-chain entirely out of LDS so the
// load->wmma dependency chain sees ~40-cycle DS latency instead of L2/HBM.
// ---------------------------------------------------------------------------
__global__ void __launch_bounds__(CONT_THREADS, 1)
cont_partial_kernel(const float* __restrict__ proj,
                    const int* __restrict__ aff,
                    const int* __restrict__ inst,
                    float* __restrict__ contP) {
    extern __shared__ float smem[];
    float* ldsProj = smem + L_PROJ0;
    float* sim     = smem + L_SIM0;          // [16][SIMSTR]
    int*   aff_sh  = (int*)(smem + L_AFF0);
    int*   inst_sh = (int*)(smem + L_INST0);
    float* pmax    = smem + L_PMAX0;         // [16][16]
    float* pse     = smem + L_PSE0;
    float* ppsim   = smem + L_PPS0;
    float* ppcnt   = smem + L_PPC0;
    float* rmax    = smem + L_RMAX0;
    float* rowwl   = smem + L_WL0;
    float* rowps   = smem + L_PS0;
    float* rowcn   = smem + L_CN0;

    const int tid  = threadIdx.x;
    const int lane = tid & 31;
    const int wave = tid >> 5;               // 0..7 (wave32)
    const int rowBase = blockIdx.x * 16;

    aff_sh[tid]  = aff[tid];
    inst_sh[tid] = inst[tid];

    // ---- stage proj[256][128] -> ldsProj[256][PL] (row-padded), coalesced --
#pragma unroll
    for (int k = 0; k < 32; ++k) {
        int g   = k * CONT_THREADS + tid;    // float4 chunk id (8192 total)
        int row = g >> 5;                    // 32 float4 per source row
        int c4  = g & 31;
        int dst = row * PL + c4 * 4;         // float offset, 16B-aligned
#if HAVE_ASYNC_LDS
        __builtin_amdgcn_global_load_async_to_lds_b128(
            (v4i*)(proj + (size_t)g * 4),    // global src (AS inferred from arg)
            (v4i*)(ldsProj + dst),           // LDS dst   (AS inferred from smem)
            0, 0);
#else
        *(v4f*)(ldsProj + dst) = *(const v4f*)(proj + (size_t)g * 4);
#endif
    }
#if HAVE_ASYNC_LDS
    asm volatile("s_wait_asynccnt 0x0" ::: "memory");
#endif
    __syncthreads();

    const int ml   = lane & 15;              // row (A) / col (B) within tile
    const int half = lane >> 4;              // lane half selects K pair

    // A fragments for this row stripe from LDS: 32 x v2f in registers
    const float* aP = ldsProj + (rowBase + ml) * PL + 2 * half;
    v2f afrag[32];
#pragma unroll
    for (int k = 0; k < 32; ++k) afrag[k] = *(const v2f*)(aP + 4 * k);

    // 2 column tiles per wave
    for (int ct = wave; ct < 16; ct += 8) {
        const int colBase = ct * 16;
        const float* bP = ldsProj + (colBase + ml) * PL + 2 * half;
        v2f bfrag[32];
#pragma unroll
        for (int k = 0; k < 32; ++k) bfrag[k] = *(const v2f*)(bP + 4 * k);

        v8f acc = {};
#pragma unroll
        for (int k = 0; k < 32; ++k) {
            acc = __builtin_amdgcn_wmma_f32_16x16x4_f32(
                /*neg_a=*/false, afrag[k], /*neg_b=*/false, bfrag[k],
                /*c_mod=*/(short)0, acc, /*reuse_a=*/false, /*reuse_b=*/false);
        }
        // C/D layout: VGPR r, lane L -> row r + 8*(L/16), col L%16
#pragma unroll
        for (int r = 0; r < 8; ++r) {
            sim[(r + 8 * half) * SIMSTR + colBase + ml] = acc[r] * TEMP_INV;
        }
    }
    __syncthreads();

    // Parallel per-row logsumexp (diag excluded) + positive-pair sums.
    // 16 threads per row, 16 columns each; fixed-order -> deterministic.
    const int r   = tid >> 4;
    const int seg = tid & 15;
    const int i   = rowBase + r;
    const int c0  = seg * 16;

    float lmax = -__builtin_inff();
#pragma unroll
    for (int c = 0; c < 16; ++c) {
        int j = c0 + c;
        if (j != i) lmax = fmaxf(lmax, sim[r * SIMSTR + j]);
    }
    pmax[r * 16 + seg] = lmax;
    __syncthreads();

    if (seg == 0) {
        float m = pmax[r * 16];
#pragma unroll
        for (int q = 1; q < 16; ++q) m = fmaxf(m, pmax[r * 16 + q]);
        rmax[r] = m;
    }
    __syncthreads();

    const float m  = rmax[r];
    const int   ai = aff_sh[i];
    const int   ii = inst_sh[i];
    float se = 0.0f, psim = 0.0f;
    int pcnt = 0;
#pragma unroll
    for (int c = 0; c < 16; ++c) {
        int j = c0 + c;
        float s = sim[r * SIMSTR + j];
        if (j != i) se += expf(s - m);
        bool pos = (aff_sh[j] == ai) && (inst_sh[j] != ii);  // diag auto-excluded
        if (pos) { psim += s; ++pcnt; }
    }
    pse[r * 16 + seg]   = se;
    ppsim[r * 16 + seg] = psim;
    ppcnt[r * 16 + seg] = (float)pcnt;
    __syncthreads();

    if (seg == 0) {
        float seT = 0.0f, psT = 0.0f, pcT = 0.0f;
#pragma unroll
        for (int q = 0; q < 16; ++q) {
            seT += pse[r * 16 + q]; psT += ppsim[r * 16 + q]; pcT += ppcnt[r * 16 + q];
        }
        float lse = m + logf(seT);
        rowwl[r] = lse * pcT;
        rowps[r] = psT;
        rowcn[r] = pcT;
    }
    __syncthreads();

    if (tid == 0) {
        float a = 0.0f, b = 0.0f, c = 0.0f;
#pragma unroll
        for (int q = 0; q < 16; ++q) { a += rowwl[q]; b += rowps[q]; c += rowcn[q]; }
        contP[blockIdx.x * 3 + 0] = a;
        contP[blockIdx.x * 3 + 1] = b;
        contP[blockIdx.x * 3 + 2] = c;
    }
}

// ---------------------------------------------------------------------------
// Kernel 3: deterministic parallel finalize -> out[3] = [total, seg, cont]
// ---------------------------------------------------------------------------
__global__ void __launch_bounds__(FIN_THREADS)
finalize_kernel(const float* __restrict__ segP,
                const float* __restrict__ contP,
                float* __restrict__ out) {
    __shared__ double rfs[FIN_THREADS], rsp[FIN_THREADS],
                      rst[FIN_THREADS], rspt[FIN_THREADS];
    const int t = threadIdx.x;

    double fs = 0.0, sp = 0.0, st = 0.0, spt = 0.0;
#pragma unroll
    for (int b = t; b < SEG_BLOCKS; b += FIN_THREADS) {
        fs  += (double)segP[4 * b + 0];
        sp  += (double)segP[4 * b + 1];
        st  += (double)segP[4 * b + 2];
        spt += (double)segP[4 * b + 3];
    }
    rfs[t] = fs; rsp[t] = sp; rst[t] = st; rspt[t] = spt;
    __syncthreads();
#pragma unroll
    for (int s = FIN_THREADS / 2; s > 0; s >>= 1) {
        if (t < s) {
            rfs[t] += rfs[t + s]; rsp[t] += rsp[t + s];
            rst[t] += rst[t + s]; rspt[t] += rspt[t + s];
        }
        __syncthreads();
    }

    if (t == 0) {
        const float M = (float)M_TOT;
        float focal = (float)(rfs[0] / (double)M_TOT);
        float spF = (float)rsp[0], stF = (float)rst[0], sptF = (float)rspt[0];

        float dice_pos = (2.0f * sptF + SMOOTH) / (spF + stF + SMOOTH);
        float inn = M - spF - stF + sptF;
        float cn  = 2.0f * M - spF - stF;
        float dice_neg = (2.0f * inn + SMOOTH) / (cn + SMOOTH);
        float dice = (1.0f - dice_pos) + (1.0f - dice_neg);
        float seg = 0.5f * focal + 0.5f * dice;

        double wl = 0.0, ps = 0.0, cnt = 0.0;
        for (int g = 0; g < CONT_GROUPS; ++g) {
            wl  += (double)contP[3 * g + 0];
            ps  += (double)contP[3 * g + 1];
            cnt += (double)contP[3 * g + 2];
        }
        float cont = (cnt > 0.0) ? (float)((wl - ps) / cnt) : 0.0f;

        float total = 1.0f * seg + 0.5f * cont;
        out[0] = total;
        out[1] = seg;
        out[2] = cont;
    }
}

// ---------------------------------------------------------------------------
extern "C" void kernel_launch(void* const* d_in, const int* in_sizes, int n_in,
                              void* d_out, int out_size, void* d_ws, size_t ws_size,
                              hipStream_t stream) {
    const float* logits = (const float*)d_in[0];   // (B, N, 1) f32
    const int*   gt     = (const int*)d_in[1];     // (B, N, 1) i32
    const float* proj   = (const float*)d_in[2];   // (B, P)    f32
    const int*   aff    = (const int*)d_in[3];     // (B,)      i32
    const int*   inst   = (const int*)d_in[4];     // (B,)      i32
    float* out = (float*)d_out;

    float* segP  = (float*)d_ws;                   // SEG_BLOCKS * 4 floats
    float* contP = segP + SEG_BLOCKS * 4;          // CONT_GROUPS * 3 floats
    // Every used ws slot is overwritten each call -> no init needed,
    // no atomics, fixed-order trees -> bitwise deterministic.

    seg_partial_kernel<<<SEG_BLOCKS, SEG_THREADS, 0, stream>>>(logits, gt, segP);
    cont_partial_kernel<<<CONT_GROUPS, CONT_THREADS, CONT_SMEM_BYTES, stream>>>(proj, aff, inst, contP);
    finalize_kernel<<<1, FIN_THREADS, 0, stream>>>(segP, contP, out);
}